// Projector_64278480552470
// MI455X (gfx1250) — compile-verified
//
#include <hip/hip_runtime.h>
#include <hip/hip_bf16.h>
#include <float.h>

typedef __attribute__((ext_vector_type(2))) float v2f;
typedef __attribute__((ext_vector_type(8))) float v8f;

#define NF 128  // feature dim of state

// ---------------------------------------------------------------------------
// Kernel 1: pack source mesh into WMMA-A-friendly records:
//   packedA[m] = { x, y, z, x^2+y^2+z^2 }
// Lane m of the main kernel then loads a single 8B chunk (K0,K1 or K2,K3).
// ---------------------------------------------------------------------------
__global__ void pack_src_kernel(const float* __restrict__ src,
                                float4* __restrict__ packedA, int n) {
    int m = blockIdx.x * blockDim.x + threadIdx.x;
    if (m >= n) return;
    float x = src[m * 3 + 0], y = src[m * 3 + 1], z = src[m * 3 + 2];
    packedA[m] = make_float4(x, y, z, x * x + y * y + z * z);
}

// Branch-free insert of (k,i) into ascending top-3 (k0<=k1<=k2).
__device__ __forceinline__ void top3_insert(float k, int i,
                                            float& k0, float& k1, float& k2,
                                            int& i0, int& i1, int& i2) {
    bool lt2 = k < k2;
    bool lt1 = k < k1;
    bool lt0 = k < k0;
    float nk2 = lt1 ? k1 : (lt2 ? k : k2);
    int   ni2 = lt1 ? i1 : (lt2 ? i : i2);
    float nk1 = lt0 ? k0 : (lt1 ? k : k1);
    int   ni1 = lt0 ? i0 : (lt1 ? i : i1);
    float nk0 = lt0 ? k  : k0;
    int   ni0 = lt0 ? i  : i0;
    k0 = nk0; k1 = nk1; k2 = nk2;
    i0 = ni0; i1 = ni1; i2 = ni2;
}

// Lexicographic (key, idx) insert — used for the cross-half merge so both
// lanes of a pair deterministically produce the identical list (stable top-k).
__device__ __forceinline__ void top3_insert_lex(float k, int i,
                                                float& k0, float& k1, float& k2,
                                                int& i0, int& i1, int& i2) {
    bool lt2 = (k < k2) || (k == k2 && i < i2);
    bool lt1 = (k < k1) || (k == k1 && i < i1);
    bool lt0 = (k < k0) || (k == k0 && i < i0);
    float nk2 = lt1 ? k1 : (lt2 ? k : k2);
    int   ni2 = lt1 ? i1 : (lt2 ? i : i2);
    float nk1 = lt0 ? k0 : (lt1 ? k : k1);
    int   ni1 = lt0 ? i0 : (lt1 ? i : i1);
    float nk0 = lt0 ? k  : k0;
    int   ni0 = lt0 ? i  : i0;
    k0 = nk0; k1 = nk1; k2 = nk2;
    i0 = ni0; i1 = ni1; i2 = ni2;
}

// ---------------------------------------------------------------------------
// Kernel 2: one wave (32 lanes) per 16-target tile.
//   key(m,n) = |s_m|^2 - 2*(s_m . t_n)   via V_WMMA_F32_16X16X4_F32
//   (A: 16 sources x {x,y,z,|s|^2};  B: {-2x,-2y,-2z,1} x 16 targets)
// Each lane owns target n = lane%16 and scans 8 source rows per WMMA,
// maintaining a top-3. The A-tile load is software-pipelined one tile ahead,
// and the insert chain is skipped wave-uniformly when no candidate in the
// tile can enter any lane's top-3 (exact: strict-< insert would be a no-op).
// Lane pairs (n, n+16) merge via shuffles, then the wave cooperatively
// gathers+averages 3 state rows per target.
// ---------------------------------------------------------------------------
__global__ __launch_bounds__(32)
void knn_project_kernel(const float* __restrict__ packedA,
                        const float* __restrict__ tgt,
                        const float* __restrict__ state,
                        float* __restrict__ out,
                        int nsTiles) {
    const int lane   = threadIdx.x;       // 0..31
    const int laneLo = lane & 15;
    const bool hiH   = lane >= 16;
    const int nBase  = blockIdx.x * 16;
    const int n      = nBase + laneLo;    // this lane's target column

    // B operand (loop-invariant): lane holds column n.
    // lanes 0-15: {K0,K1} = {-2x, -2y}; lanes 16-31: {K2,K3} = {-2z, 1}
    v2f b;
    if (!hiH) {
        b.x = -2.0f * tgt[n * 3 + 0];
        b.y = -2.0f * tgt[n * 3 + 1];
    } else {
        b.x = -2.0f * tgt[n * 3 + 2];
        b.y = 1.0f;
    }

    float k0 = FLT_MAX, k1 = FLT_MAX, k2 = FLT_MAX;
    int   i0 = 0, i1 = 0, i2 = 0;
    const int hiOff = hiH ? 8 : 0;

    // Per-lane A pointer: row (tile*16 + laneLo), K-half by lane half.
    const float* aPtr = packedA + (size_t)laneLo * 4 + (hiH ? 2 : 0);

    // Software pipeline: keep the next tile's A load in flight across the
    // WMMA + selection of the current tile.
    v2f a = *(const v2f*)aPtr;
    for (int tile = 0; tile < nsTiles; ++tile) {
        const int nextTile = (tile + 1 < nsTiles) ? tile + 1 : tile;
        v2f aNext = *(const v2f*)(aPtr + (size_t)nextTile * 64);

        v8f c = {0.f, 0.f, 0.f, 0.f, 0.f, 0.f, 0.f, 0.f};
        c = __builtin_amdgcn_wmma_f32_16x16x4_f32(
                /*neg_a=*/false, a, /*neg_b=*/false, b,
                /*c_mod=*/(short)0, c, /*reuse_a=*/false, /*reuse_b=*/false);

        // Wave-uniform fast path: if no candidate in this tile beats any
        // lane's current k2, the strict-< insert chain is a guaranteed no-op.
        float m01 = fminf(c[0], c[1]);
        float m23 = fminf(c[2], c[3]);
        float m45 = fminf(c[4], c[5]);
        float m67 = fminf(c[6], c[7]);
        float min8 = fminf(fminf(m01, m23), fminf(m45, m67));
        if (__ballot(min8 < k2) != 0ull) {
            const int iBase = tile * 16 + hiOff;
#pragma unroll
            for (int v = 0; v < 8; ++v) {
                top3_insert(c[v], iBase + v, k0, k1, k2, i0, i1, i2);
            }
        }
        a = aNext;
    }

    // Merge the two halves (same target, disjoint source rows).
    float fk0 = __shfl_xor(k0, 16, 32);
    float fk1 = __shfl_xor(k1, 16, 32);
    float fk2 = __shfl_xor(k2, 16, 32);
    int   fi0 = __shfl_xor(i0, 16, 32);
    int   fi1 = __shfl_xor(i1, 16, 32);
    int   fi2 = __shfl_xor(i2, 16, 32);
    top3_insert_lex(fk0, fi0, k0, k1, k2, i0, i1, i2);
    top3_insert_lex(fk1, fi1, k0, k1, k2, i0, i1, i2);
    top3_insert_lex(fk2, fi2, k0, k1, k2, i0, i1, i2);

    // Gather + mean: broadcast each target's 3 indices, all 32 lanes sweep
    // the 128-float feature rows coalesced (4 iters of 32 lanes).
    const float scale = 1.0f / 3.0f;
    for (int t = 0; t < 16; ++t) {
        int a0 = __shfl(i0, t, 32);
        int a1 = __shfl(i1, t, 32);
        int a2 = __shfl(i2, t, 32);
        const float* s0 = state + (size_t)a0 * NF;
        const float* s1 = state + (size_t)a1 * NF;
        const float* s2 = state + (size_t)a2 * NF;
        float* op = out + (size_t)(nBase + t) * NF;
        for (int f = lane; f < NF; f += 32) {
            op[f] = (s0[f] + s1[f] + s2[f]) * scale;
        }
    }
}

extern "C" void kernel_launch(void* const* d_in, const int* in_sizes, int n_in,
                              void* d_out, int out_size, void* d_ws, size_t ws_size,
                              hipStream_t stream) {
    const float* src   = (const float*)d_in[0];  // [Ns,3]
    const float* tgt   = (const float*)d_in[1];  // [Nt,3]
    const float* state = (const float*)d_in[2];  // [Ns,128]
    float*       out   = (float*)d_out;          // [Nt,128]

    const int ns = in_sizes[0] / 3;              // 16384
    const int nt = in_sizes[1] / 3;              // 16384

    float4* packedA = (float4*)d_ws;             // ns * 16 bytes

    pack_src_kernel<<<(ns + 255) / 256, 256, 0, stream>>>(src, packedA, ns);
    knn_project_kernel<<<nt / 16, 32, 0, stream>>>(
        (const float*)packedA, tgt, state, out, ns / 16);
}